// LongMemoryBank_63187558859376
// MI455X (gfx1250) — compile-verified
//
#include <hip/hip_runtime.h>
#include <math.h>

// ---------------------------------------------------------------------------
// LongMemoryBank compress: concat [16,8192,512]+[16,32,512] then 32x merge of
// the adjacent pair with minimal summed L2 norm. Three phases:
//   1) WMMA Gram-diagonal pass -> per-slot L2 norms (memory bound)
//   2) per-batch merge simulation in LDS (tiny)
//   3) vectorized gather of the surviving/merged rows (memory bound)
// ---------------------------------------------------------------------------

typedef __attribute__((ext_vector_type(2))) float v2f;
typedef __attribute__((ext_vector_type(8))) float v8f;

#define BATCH 16
#define S_BANK 8192
#define S_REF 32
#define S_TOT (S_BANK + S_REF)      // 8224
#define DIM 512
#define N_MERGE (S_TOT - S_BANK)    // 32 merge steps
#define TILES_PER_B (S_TOT / 16)    // 514 WMMA tiles per batch

__device__ __forceinline__ const float* slot_ptr(const float* bank,
                                                 const float* refresh,
                                                 const float* scratch,
                                                 int b, int id) {
  if (id < S_BANK)
    return bank + ((size_t)b * S_BANK + id) * DIM;
  else if (id < S_TOT)
    return refresh + ((size_t)b * S_REF + (id - S_BANK)) * DIM;
  else
    return scratch + ((size_t)b * N_MERGE + (id - S_TOT)) * DIM;
}

// ---------------------------------------------------------------------------
// Phase 1: per-slot L2 norms (sqrt of Gram diagonal) via V_WMMA_F32_16X16X4_F32.
// One wave32 handles 16 consecutive slots. A-layout (ISA 7.12.2, 32-bit A
// 16x4): lanes 0-15 hold K=k0,k0+1 for M=lane; lanes 16-31 hold K=k0+2,k0+3
// for M=lane-16. With B = A^T the register images coincide, so the same v2f
// feeds SRC0 and SRC1; the accumulated C diagonal is sum_k x[m,k]^2.
// ---------------------------------------------------------------------------
__global__ __launch_bounds__(256)
void imp_wmma_kernel(const float* __restrict__ bank,
                     const float* __restrict__ refresh,
                     float* __restrict__ imp) {
  const int wave = blockIdx.x * 8 + (threadIdx.x >> 5);
  const int lane = threadIdx.x & 31;
  const int b = wave / TILES_PER_B;
  const int tile = wave - b * TILES_PER_B;
  const int slot0 = tile * 16;
  const int slot = slot0 + (lane & 15);

  const float* row = (slot < S_BANK)
      ? bank + ((size_t)b * S_BANK + slot) * DIM
      : refresh + ((size_t)b * S_REF + (slot - S_BANK)) * DIM;
  // lanes 0-15 read floats {k0,k0+1}; lanes 16-31 read {k0+2,k0+3}
  const float2* p2 = (const float2*)(row + ((lane >> 4) << 1));

  v8f acc = {};
#pragma unroll 4
  for (int k = 0; k < DIM / 4; ++k) {
    float2 d = p2[k * 2];
    v2f a;
    a[0] = d.x;
    a[1] = d.y;
    acc = __builtin_amdgcn_wmma_f32_16x16x4_f32(false, a, false, a,
                                                (short)0, acc, false, false);
  }

  // C/D layout: VGPR r -> M=r (lanes 0-15, N=lane) and M=r+8 (lanes 16-31,
  // N=lane-16). Diagonal element n: n<8 at (lane n, reg n); n>=8 at
  // (lane n+16, reg n-8). Store imp = sqrt(sumsq) to match the reference.
  float* impRow = imp + (size_t)b * S_TOT + slot0;
#pragma unroll
  for (int r = 0; r < 8; ++r) {
    if (lane == r)       impRow[r]     = sqrtf(acc[r]);
    if (lane == r + 24)  impRow[r + 8] = sqrtf(acc[r]);
  }
}

// ---------------------------------------------------------------------------
// Phase 2: one block per batch row. Tombstone simulation of 32 merges.
// LDS: 8224 f32 norms + 8224 u16 ids + reductions  (~52 KB, fits 64 KB).
// Argmin key = (float_bits(score) << 32) | j  -> lexicographic (score, j)
// minimum == jnp.argmin first-occurrence semantics (scores >= 0).
// ---------------------------------------------------------------------------
__global__ __launch_bounds__(256)
void merge_sim_kernel(const float* __restrict__ bank,
                      const float* __restrict__ refresh,
                      const float* __restrict__ imp,
                      float* __restrict__ scratch,
                      int* __restrict__ mapOut) {
  __shared__ float sImp[S_TOT];
  __shared__ unsigned short sCur[S_TOT];   // source id, 0xFFFF = deleted
  __shared__ unsigned long long rkey[256];
  __shared__ float rsum[256];
  __shared__ int sInv[N_MERGE];

  const int b = blockIdx.x;
  const int tid = threadIdx.x;

  for (int i = tid; i < S_TOT; i += 256) {
    sImp[i] = imp[(size_t)b * S_TOT + i];
    sCur[i] = (unsigned short)i;
  }
  __syncthreads();

  for (int t = 0; t < N_MERGE; ++t) {
    // ---- argmin over adjacent (valid) pairs ----
    unsigned long long best = ~0ull;
    for (int j = tid; j < S_TOT - 1; j += 256) {
      if (sCur[j] == 0xFFFFu) continue;
      int k = j + 1;
      while (k < S_TOT && sCur[k] == 0xFFFFu) ++k;   // <=32 tombstones total
      if (k >= S_TOT) continue;                      // j is last valid slot
      float sc = sImp[j] + sImp[k];
      unsigned long long key =
          ((unsigned long long)__float_as_uint(sc) << 32) | (unsigned)j;
      if (key < best) best = key;
    }
    rkey[tid] = best;
    __syncthreads();
    for (int s = 128; s > 0; s >>= 1) {
      if (tid < s) {
        unsigned long long o = rkey[tid + s];
        if (o < rkey[tid]) rkey[tid] = o;
      }
      __syncthreads();
    }
    const int p = (int)(rkey[0] & 0xFFFFFFFFu);
    int q = p + 1;
    while (sCur[q] == 0xFFFFu) ++q;   // identical result in every thread

    // ---- materialize merged row, compute its norm ----
    const int ia = sCur[p];
    const int ib = sCur[q];
    const float* va = slot_ptr(bank, refresh, scratch, b, ia);
    const float* vb = slot_ptr(bank, refresh, scratch, b, ib);
    float* vm = scratch + ((size_t)b * N_MERGE + t) * DIM;
    float ls = 0.f;
    for (int e = tid; e < DIM; e += 256) {
      float m = 0.5f * (va[e] + vb[e]);
      vm[e] = m;
      ls += m * m;
    }
    rsum[tid] = ls;
    __syncthreads();
    for (int s = 128; s > 0; s >>= 1) {
      if (tid < s) rsum[tid] += rsum[tid + s];
      __syncthreads();
    }
    if (tid == 0) {
      sImp[p] = sqrtf(rsum[0]);
      sCur[p] = (unsigned short)(S_TOT + t);
      sCur[q] = 0xFFFFu;
      sInv[t] = q;
    }
    __syncthreads();
  }

  // ---- compaction: out position = j - #(tombstones < j) ----
  for (int j = tid; j < S_TOT; j += 256) {
    unsigned short id = sCur[j];
    if (id == 0xFFFFu) continue;
    int pos = j;
#pragma unroll
    for (int t = 0; t < N_MERGE; ++t) pos -= (sInv[t] < j) ? 1 : 0;
    mapOut[(size_t)b * S_BANK + pos] = (int)id;
  }
}

// ---------------------------------------------------------------------------
// Phase 3: gather surviving rows. 2 rows per 256-thread block, float4 copies.
// ---------------------------------------------------------------------------
__global__ __launch_bounds__(256)
void gather_kernel(const float* __restrict__ bank,
                   const float* __restrict__ refresh,
                   const float* __restrict__ scratch,
                   const int* __restrict__ mapIn,
                   float* __restrict__ out) {
  const int gRow = blockIdx.x * 2 + (threadIdx.x >> 7);
  const int l = threadIdx.x & 127;           // 128 lanes * float4 = 512 floats
  const int b = gRow >> 13;                  // / 8192
  const int j = gRow & (S_BANK - 1);
  const int id = mapIn[(size_t)b * S_BANK + j];
  const float4* s4 = (const float4*)slot_ptr(bank, refresh, scratch, b, id);
  float4* d4 = (float4*)(out + (size_t)gRow * DIM);
  d4[l] = s4[l];
}

extern "C" void kernel_launch(void* const* d_in, const int* in_sizes, int n_in,
                              void* d_out, int out_size, void* d_ws, size_t ws_size,
                              hipStream_t stream) {
  const float* bank = (const float*)d_in[0];     // [16,8192,512] f32
  const float* refresh = (const float*)d_in[1];  // [16,32,512]  f32
  float* out = (float*)d_out;                    // [16,8192,512] f32

  // workspace layout (~2.1 MB): norms | map | merged-row scratch
  float* imp = (float*)d_ws;
  int* map = (int*)((char*)d_ws + sizeof(float) * (size_t)BATCH * S_TOT);
  float* scratch = (float*)((char*)d_ws + sizeof(float) * (size_t)BATCH * S_TOT
                                        + sizeof(int) * (size_t)BATCH * S_BANK);

  // Phase 1: 16*514 = 8224 waves, 8 waves (256 thr) per block -> 1028 blocks.
  imp_wmma_kernel<<<(BATCH * TILES_PER_B) / 8, 256, 0, stream>>>(bank, refresh, imp);
  // Phase 2: one block per batch row.
  merge_sim_kernel<<<BATCH, 256, 0, stream>>>(bank, refresh, imp, scratch, map);
  // Phase 3: 2 rows per block.
  gather_kernel<<<(BATCH * S_BANK) / 2, 256, 0, stream>>>(bank, refresh, scratch, map, out);
}